// RandHashProj_36318243455052
// MI455X (gfx1250) — compile-verified
//
#include <hip/hip_runtime.h>
#include <stdint.h>

// Problem dims (fixed by the reference).
#define OUT_FEAT 8192
#define IN_FEAT  8192
#define BATCH    4096

typedef unsigned int u32;
typedef __attribute__((ext_vector_type(4))) u32   v4u;
typedef __attribute__((ext_vector_type(4))) int   v4i;
typedef __attribute__((ext_vector_type(8))) int   v8i;
typedef __attribute__((ext_vector_type(4))) float v4f;

#if __has_builtin(__builtin_amdgcn_tensor_load_to_lds) && __has_builtin(__builtin_amdgcn_s_wait_tensorcnt)
#define USE_TDM 1
#else
#define USE_TDM 0
#endif

// ------------------------------------------------------------------
// TDM: async DMA of a contiguous f32 row chunk from global into LDS.
// Builds a D# (groups 0/1; remaining groups zero) per CDNA5 ISA:
//   g0: count=1 | lds_addr | global_addr[56:0] | type=2
//   g1: data_size=4B | tensor_dim0=n | tensor_dim1=1 | tile_dim0=n |
//       tile_dim1=1 | tensor_dim0_stride=n
// Tracked by TENSORcnt (per-wave, EXEC-ignored): issue from wave 0 only,
// publish to the workgroup with a barrier.
// ------------------------------------------------------------------
__device__ __forceinline__ void tdm_load_row_chunk(const float* gsrc,
                                                   float* lds_dst, int nelem) {
#if USE_TDM
  const unsigned long long ga = (unsigned long long)(uintptr_t)gsrc;
  // Generic LDS pointer: low 32 bits are the wave-relative LDS byte offset.
  const u32 lds_off = (u32)(uintptr_t)lds_dst;
  v4u g0;
  g0.x = 1u;                                             // count=1 (valid user D#)
  g0.y = lds_off;                                        // lds_addr
  g0.z = (u32)ga;                                        // global_addr[31:0]
  g0.w = (u32)((ga >> 32) & 0x01FFFFFFull) | (2u << 30); // addr[56:32] | type=2
  const u32 n = (u32)nelem;
  v8i g1;
  g1[0] = (int)(2u << 16);                 // data_size = 2 (4 bytes)
  g1[1] = (int)((n & 0xFFFFu) << 16);      // tensor_dim0[15:0]
  g1[2] = (int)(((n >> 16) & 0xFFFFu)      // tensor_dim0[31:16]
                | (1u << 16));             // tensor_dim1 = 1
  g1[3] = (int)((n & 0xFFFFu) << 16);      // tile_dim0 = n
  g1[4] = 1;                               // tile_dim1 = 1, tile_dim2 = 0
  g1[5] = (int)n;                          // tensor_dim0_stride[31:0]
  g1[6] = 0;                               // stride hi / dim1_stride lo
  g1[7] = 0;
  const v4i gz4 = {0, 0, 0, 0};            // groups 2/3 unused (<=2D tensor)
  const v8i gz8 = {0, 0, 0, 0, 0, 0, 0, 0};
  __builtin_amdgcn_tensor_load_to_lds(g0, g1, gz4, gz4, gz8, 0);
#else
  // Fallback (other toolchain): synchronous copy, correctness only.
  for (int i = 0; i < nelem; ++i) lds_dst[i] = gsrc[i];
#endif
}

// ------------------------------------------------------------------
// Phase 2: one block per batch row.
//   LDS: 32 KB f32 accumulators (full output row, exact f32 math)
//        + 2 x 8 KB double-buffered TDM staging of the x row.
//   Chunk-0 DMA is issued FIRST so it overlaps the accumulator zeroing.
//   Per feature: one LDS read + one ds_add_f32 scatter into acc.
//   48 KB LDS/block -> 6 blocks per 320 KB WGP.
// ------------------------------------------------------------------
#define CHUNK 2048
#define NCH   (IN_FEAT / CHUNK)

__global__ __launch_bounds__(256) void scatter_rows_kernel(
    const float* __restrict__ x, const int* __restrict__ sel,
    float* __restrict__ out) {
  __shared__ float acc[OUT_FEAT];     // 32 KB accumulators
  __shared__ float xbuf[2][CHUNK];    // 16 KB staging (double buffered)

  const int b = blockIdx.x;
  const float* xrow = x + (size_t)b * IN_FEAT;

  // Kick off the chunk-0 DMA before anything else touches memory.
  if (threadIdx.x == 0) tdm_load_row_chunk(xrow, xbuf[0], CHUNK);

  // Zero the accumulators while the DMA is in flight (disjoint LDS ranges).
  for (int i = threadIdx.x; i < OUT_FEAT; i += 256) acc[i] = 0.0f;
  __syncthreads();  // accumulators zeroed before any atomics

  for (int ch = 0; ch < NCH; ++ch) {
    if (threadIdx.x == 0) {
      if (ch + 1 < NCH) {
        tdm_load_row_chunk(xrow + (size_t)(ch + 1) * CHUNK,
                           xbuf[(ch + 1) & 1], CHUNK);
#if USE_TDM
        // TDM completes in-order: <=1 outstanding => chunk `ch` landed.
        __builtin_amdgcn_s_wait_tensorcnt(1);
#endif
      } else {
#if USE_TDM
        __builtin_amdgcn_s_wait_tensorcnt(0);
#endif
      }
    }
    __syncthreads();  // publish chunk `ch` to all 8 waves

    const float* xb = xbuf[ch & 1];
    const int fbase = ch * CHUNK;
    for (int i = threadIdx.x; i < CHUNK; i += 256) {
      const int s = sel[fbase + i];       // packed (col<<1)|neg, hot in L2
      const float v = xb[i];              // ds_load from staged chunk
      atomicAdd(&acc[s >> 1], (s & 1) ? -v : v);  // ds_add_f32
    }
    __syncthreads();  // buffer free before TDM reuses it (2 iters later)
  }

  // Stream out the finished row (non-temporal b128 stores).
  v4f* o4 = (v4f*)(out + (size_t)b * OUT_FEAT);
  const v4f* a4 = (const v4f*)acc;
  for (int i = threadIdx.x; i < OUT_FEAT / 4; i += 256)
    __builtin_nontemporal_store(a4[i], o4 + i);
}

// ------------------------------------------------------------------
// Phase 1: decode the sparse projection matrix.
// proj has exactly one nonzero (+1/-1) per row; pack (col<<1)|sign_bit.
// 256 MB streamed once -> non-temporal loads to keep L2 for phase 2.
// ------------------------------------------------------------------
__global__ __launch_bounds__(256) void decode_proj_kernel(
    const float* __restrict__ proj, int* __restrict__ sel) {
  const int r = blockIdx.x;
  const v4f* p = (const v4f*)(proj + (size_t)r * OUT_FEAT);
  for (int i = threadIdx.x; i < OUT_FEAT / 4; i += 256) {
    v4f v = __builtin_nontemporal_load(p + i);
    const int c = i * 4;
    if (v.x != 0.0f) sel[r] = ((c + 0) << 1) | (v.x < 0.0f ? 1 : 0);
    if (v.y != 0.0f) sel[r] = ((c + 1) << 1) | (v.y < 0.0f ? 1 : 0);
    if (v.z != 0.0f) sel[r] = ((c + 2) << 1) | (v.z < 0.0f ? 1 : 0);
    if (v.w != 0.0f) sel[r] = ((c + 3) << 1) | (v.w < 0.0f ? 1 : 0);
  }
}

// ------------------------------------------------------------------
extern "C" void kernel_launch(void* const* d_in, const int* in_sizes, int n_in,
                              void* d_out, int out_size, void* d_ws,
                              size_t ws_size, hipStream_t stream) {
  (void)in_sizes; (void)n_in; (void)out_size; (void)ws_size;
  const float* x    = (const float*)d_in[0];   // (4096, 8192) f32
  const float* proj = (const float*)d_in[1];   // (8192, 8192) f32
  float* out = (float*)d_out;                  // (4096, 8192) f32
  int* sel = (int*)d_ws;                       // 8192 ints = 32 KB scratch

  decode_proj_kernel<<<IN_FEAT, 256, 0, stream>>>(proj, sel);
  scatter_rows_kernel<<<BATCH, 256, 0, stream>>>(x, sel, out);
}